// Activation1d_50525995270785
// MI455X (gfx1250) — compile-verified
//
#include <hip/hip_runtime.h>
#include <hip/hip_bf16.h>
#include <stdint.h>

// out[j] = 0.5*( silu(0.25*x[j-1] + 0.75*x[j]) + silu(0.75*x[j] + 0.25*x[j+1]) )
// (edge-clamped along T). HBM-bound stencil: 512 MB traffic -> ~22us floor at
// 23.3 TB/s. TDM (tensor_load_to_lds) stages each block's row tile into LDS;
// SiLU uses v_tanh_f32 / v_exp_f32+v_rcp_f32 to keep VALU below the BW floor;
// outputs leave via nontemporal b128 stores.

#define TDIM      8192     // T from the reference
#define TILE      2048     // elements of one row per block
#define NTHREADS  256
#define PER_TH    8        // TILE / NTHREADS

typedef unsigned int v4u __attribute__((ext_vector_type(4)));
typedef int          v8i __attribute__((ext_vector_type(8)));
typedef int          v4i __attribute__((ext_vector_type(4)));
typedef float        v4f __attribute__((ext_vector_type(4)));

// Returns 0.5f * silu(x). 0.5*silu(x) = q*tanh(x/2) + q with q = x/4.
__device__ __forceinline__ float half_silu(float x) {
#if __has_builtin(__builtin_amdgcn_tanhf)
    float h = 0.5f * x;
    float t = __builtin_amdgcn_tanhf(h);     // v_tanh_f32 (1 trans)
    float q = 0.5f * h;
    return __builtin_fmaf(q, t, q);
#elif __has_builtin(__builtin_amdgcn_tanh_f32)
    float h = 0.5f * x;
    float t = __builtin_amdgcn_tanh_f32(h);
    float q = 0.5f * h;
    return __builtin_fmaf(q, t, q);
#else
    // 0.5*x * 1/(1 + exp(-x)); exp(-x) = exp2(-log2(e)*x)
    float e = __builtin_amdgcn_exp2f(-1.44269504f * x);  // v_exp_f32
    float r = __builtin_amdgcn_rcpf(1.0f + e);           // v_rcp_f32
    return (0.5f * x) * r;
#endif
}

__global__ __launch_bounds__(NTHREADS)
void act1d_tdm_kernel(const float* __restrict__ x, float* __restrict__ out) {
    // s[i] mirrors x[row, t0 - 1 + i] for i in [0, TILE+1] (edge-clamped halos)
    __shared__ __align__(16) float s[TILE + 2];

    const int t0  = blockIdx.x * TILE;          // tile start along T
    const int row = blockIdx.y;                 // flattened (b, c)
    const float* __restrict__ xrow = x   + (size_t)row * TDIM;
    float*       __restrict__ orow = out + (size_t)row * TDIM;

    // ---------------- TDM: DMA xrow[t0 .. t0+TILE) -> s[1 .. TILE] ----------
    if (threadIdx.x < 32) {   // one DMA issued by wave 0 (EXEC ignored by TDM)
        const unsigned lds_base = (unsigned)(unsigned long long)
            (__attribute__((address_space(3))) float*)(&s[0]);
        const unsigned long long ga =
            (unsigned long long)(uintptr_t)(xrow + t0);

        // D# group0: count=1 | lds_addr | global_addr[56:0] | type=2
        v4u g0 = {
            1u,                                            // count=1, user mode
            lds_base + 4u,                                 // dest = &s[1]
            (unsigned)(ga & 0xFFFFFFFFull),                // global_addr lo
            (((unsigned)(ga >> 32)) & 0x01FFFFFFu) | 0x80000000u  // hi | type=2
        };

        // D# group1: data_size=4B, tensor_dim0=TILE, tensor_dim1=1,
        //            tile_dim0=TILE (1-D tile: tile_dim1=tile_dim2=0)
        v8i g1 = {
            (int)(2u << 16),                         // wg_mask=0, data_size=2 (4B)
            (int)(((unsigned)TILE & 0xFFFFu) << 16), // tensor_dim0[15:0] << 16
            (int)(((unsigned)TILE >> 16) | (1u << 16)), // tdim0[31:16] | tdim1.lo=1
            (int)(((unsigned)TILE & 0xFFFFu) << 16), // tdim1.hi=0 | tile_dim0 << 16
            0,                                       // tile_dim1=0, tile_dim2=0
            (int)TILE,                               // tensor_dim0_stride lo32
            0, 0                                     // stride hi / dim1_stride = 0
        };

        // Groups 2/3: higher dims inert (dims=1, tiles=0)
        v4i g2 = { 1, 1, 0, 0 };                 // tensor_dim2=1, tensor_dim3=1
        v4i g3 = { 0, (int)(1u << 16), 0, 0 };   // tensor_dim4=1

        // 6-arg toolchain form (clang-23/therock): extra int32x8 operand.
        v8i gx = { 0, 0, 0, 0, 0, 0, 0, 0 };
        __builtin_amdgcn_tensor_load_to_lds(g0, g1, g2, g3, gx, /*cpol=*/0);
    }

    // Halo elements (clamped to the row), split across two waves; addresses
    // are disjoint from the TDM destination range -> no ordering hazard.
    if (threadIdx.x == 32) {
        int li = t0 - 1;        if (li < 0)         li = 0;
        s[0]        = xrow[li];
    }
    if (threadIdx.x == 64) {
        int ri = t0 + TILE;     if (ri > TDIM - 1)  ri = TDIM - 1;
        s[TILE + 1] = xrow[ri];
    }

    __builtin_amdgcn_s_wait_tensorcnt(0);   // issuing wave: DMA landed in LDS
    __syncthreads();                        // publish LDS to all waves

    // ---------------- Stencil + SiLU from LDS, nontemporal b128 stores ------
    const int l0 = threadIdx.x * PER_TH;    // 32B-aligned window into s
    float v[PER_TH + 2];
    {
        const v4f* sv = (const v4f*)(&s[l0]);
        v4f a = sv[0];
        v4f b = sv[1];
        v[0] = a.x; v[1] = a.y; v[2] = a.z; v[3] = a.w;
        v[4] = b.x; v[5] = b.y; v[6] = b.z; v[7] = b.w;
        v[8] = s[l0 + 8];
        v[9] = s[l0 + 9];
    }

    float r[PER_TH];
#pragma unroll
    for (int k = 0; k < PER_TH; ++k) {
        // x[j-1]=v[k], x[j]=v[k+1], x[j+1]=v[k+2]
        float lo = __builtin_fmaf(0.25f, v[k],     0.75f * v[k + 1]);
        float hi = __builtin_fmaf(0.25f, v[k + 2], 0.75f * v[k + 1]);
        r[k] = half_silu(lo) + half_silu(hi);
    }

    v4f* ov = (v4f*)(orow + t0 + l0);
    v4f r0 = { r[0], r[1], r[2], r[3] };
    v4f r1 = { r[4], r[5], r[6], r[7] };
    __builtin_nontemporal_store(r0, ov);        // global_store_b128 ... nt
    __builtin_nontemporal_store(r1, ov + 1);
}

extern "C" void kernel_launch(void* const* d_in, const int* in_sizes, int n_in,
                              void* d_out, int out_size, void* d_ws, size_t ws_size,
                              hipStream_t stream) {
    (void)n_in; (void)d_ws; (void)ws_size; (void)out_size;
    const float* x = (const float*)d_in[0];
    float* out     = (float*)d_out;

    const long long n   = (long long)in_sizes[0];   // B*C*T
    const int rows      = (int)(n / TDIM);          // B*C = 8192
    dim3 grid(TDIM / TILE, rows);                   // (4, 8192) blocks
    act1d_tdm_kernel<<<grid, NTHREADS, 0, stream>>>(x, out);
}